// DTree_84061099917446
// MI455X (gfx1250) — compile-verified
//
#include <hip/hip_runtime.h>
#include <hip/hip_bf16.h>

// Soft decision tree forward for MI455X (gfx1250, wave32, WMMA).
//
// out[n] = sum_leaf ( prod_d pathfactor(g[n, node(d,leaf)]) ) * cls[leaf]
// g = sigmoid(x @ W^T - c),  W = relu(FI),  c = sum_k W*sigmoid(FS)
//
// Phase 1: x @ W^T via V_WMMA_F32_16X16X4_F32 (fp32-exact), sigmoid epilogue
//          using native v_exp_f32 + v_rcp_f32 (no IEEE-division sequence),
//          result transposed into LDS g[row][node] (stride 257 = conflict-free).
// Phase 2: one (row, half) per thread; leaf products share path prefixes
//          (only the ctz-suffix of the depth chain is recomputed per leaf).

#define FDIM   32
#define NODES  255
#define LEAVES 256
#define NW     4                  // waves per block
#define ROWS_PER_BLOCK (16 * NW)  // 64 rows
#define GSTRIDE 257               // padded row stride for g in LDS

typedef float v2f __attribute__((ext_vector_type(2)));
typedef float v8f __attribute__((ext_vector_type(8)));

// Fast sigmoid: 1/(1+e^-t) with e^-t via v_exp_f32 and the divide via
// v_rcp_f32 (operand >= 1, rcp is ~1 ulp -> plenty for this workload).
__device__ __forceinline__ float sigmoidf(float t) {
    return __builtin_amdgcn_rcpf(1.0f + __expf(-t));
}

__global__ __launch_bounds__(32 * NW)
void dtree_wmma_kernel(const float* __restrict__ x,
                       const float* __restrict__ fi,
                       const float* __restrict__ fs,
                       const float* __restrict__ cls,
                       float* __restrict__ out)
{
    __shared__ float Wl[LEAVES * FDIM];          // 32 KB, node row 255 zeroed
    __shared__ float cl[LEAVES];                 // 1 KB
    __shared__ float clsl[LEAVES];               // 1 KB
    __shared__ float gl[NW][16 * GSTRIDE];       // ~64.2 KB

    const int tid  = threadIdx.x;
    const int lane = tid & 31;
    const int wave = tid >> 5;

    // ---- Preload W = relu(FI); c = sum_k W * sigmoid(FS); cls ----
    for (int i = tid; i < LEAVES * FDIM; i += blockDim.x) {
        Wl[i] = (i < NODES * FDIM) ? fmaxf(fi[i], 0.0f) : 0.0f;
    }
    for (int nd = tid; nd < LEAVES; nd += blockDim.x) {
        float acc = 0.0f;
        if (nd < NODES) {
            #pragma unroll 4
            for (int k = 0; k < FDIM; ++k) {
                float w = fmaxf(fi[nd * FDIM + k], 0.0f);
                float s = sigmoidf(fs[nd * FDIM + k]);
                acc += w * s;
            }
        }
        cl[nd]   = acc;
        clsl[nd] = cls[nd];
    }
    __syncthreads();

    // ---- Phase 1: g = sigmoid(x @ W^T - c) via WMMA f32 16x16x4 ----
    const int row_base = blockIdx.x * ROWS_PER_BLOCK + wave * 16;
    const int mrow = lane & 15;        // A: M = lane % 16
    const int koff = 2 * (lane >> 4);  // A/B: K sub-offset per lane half

    // Hoist A tile: x[row_base + mrow][0..31], 8 chunks of float2 per lane
    const float* xrow = x + (size_t)(row_base + mrow) * FDIM + koff;
    v2f a[8];
    #pragma unroll
    for (int kc = 0; kc < 8; ++kc)
        a[kc] = *(const v2f*)(xrow + 4 * kc);

    #pragma unroll 1
    for (int nt = 0; nt < 16; ++nt) {
        const int node = nt * 16 + (lane & 15);      // B: N = lane % 16
        const float* wrow = &Wl[node * FDIM + koff];

        v8f acc = {};
        #pragma unroll
        for (int kc = 0; kc < 8; ++kc) {
            v2f b = *(const v2f*)(wrow + 4 * kc);
            acc = __builtin_amdgcn_wmma_f32_16x16x4_f32(
                false, a[kc], false, b, (short)0, acc, false, false);
        }

        // Epilogue: D layout -> lane holds N=node, M = r + 8*(lane>>4)
        const float cv = cl[node];
        #pragma unroll
        for (int r = 0; r < 8; ++r) {
            const int rloc = r + 8 * (lane >> 4);
            gl[wave][rloc * GSTRIDE + node] = sigmoidf(acc[r] - cv);
        }
    }
    __syncthreads();

    // ---- Phase 2: leaf-product tree walk, 2 threads per row ----
    const int row  = tid >> 1;   // 0..63 within block
    const int half = tid & 1;    // leaves [half*128, half*128+128)
    const float* grow = &gl[row >> 4][(row & 15) * GSTRIDE];

    float p[8];
    float acc = 0.0f;
    const int lf0 = half * 128;

    for (int j = 0; j < 128; ++j) {
        const int lf = lf0 + j;
        // Only depths >= dstart change between consecutive leaves.
        int dstart = 0;
        if (j != 0) {
            int t = __builtin_ctz((unsigned)lf);   // <= 7 here
            dstart = 7 - t;
        }
        #pragma unroll
        for (int d0 = 0; d0 < 8; ++d0) {
            if (d0 >= dstart) {
                const int nodei = (1 << d0) - 1 + (lf >> (8 - d0));
                const int pb    = (lf >> (7 - d0)) & 1;
                const float gv  = grow[nodei];
                const float f   = pb ? (1.0f - gv) : gv;
                p[d0] = ((d0 == 0) ? 1.0f : p[d0 - 1]) * f;
            }
        }
        acc += p[7] * clsl[lf];
    }

    // Combine the two leaf-halves of each row.
    const float other = __shfl_xor(acc, 1);
    if (half == 0) {
        out[(size_t)blockIdx.x * ROWS_PER_BLOCK + row] = acc + other;
    }
}

extern "C" void kernel_launch(void* const* d_in, const int* in_sizes, int n_in,
                              void* d_out, int out_size, void* d_ws, size_t ws_size,
                              hipStream_t stream) {
    const float* x   = (const float*)d_in[0];
    const float* fi  = (const float*)d_in[1];
    const float* fs  = (const float*)d_in[2];
    const float* cls = (const float*)d_in[3];
    float* out = (float*)d_out;

    const int rows = in_sizes[0] / FDIM;          // 262144
    const int grid = rows / ROWS_PER_BLOCK;       // 4096

    dtree_wmma_kernel<<<grid, 32 * NW, 0, stream>>>(x, fi, fs, cls, out);
}